// Attention_37752762532690
// MI455X (gfx1250) — compile-verified
//
#include <hip/hip_runtime.h>

// Problem shape (fixed by the reference): B=8, S=2048, H=1024, fp32.
#define BATCH 8
#define SEQ   2048
#define HID   1024

typedef __attribute__((ext_vector_type(2))) float v2f;
typedef __attribute__((ext_vector_type(8))) float v8f;

// Column-sum of encoder_outputs[b, :, :] over the S axis via
// V_WMMA_F32_16X16X4_F32 with an all-ones A matrix:
//   D[i,n] = sum_k A[i,k]*B[k,n] + C[i,n] = sum_k B[k,n] + C[i,n]
// Each WMMA consumes a 4(s) x 16(h) f32 tile of encoder_outputs.
// Since A is all-ones, every row of D carries the same column sum, so
// acc[0] on lane n (n = lane&15) is the reduction result for column n.
__global__ __launch_bounds__(256) void colmean_wmma_kernel(
    const float* __restrict__ enc,   // (B, S, H)
    float* __restrict__ context)     // (B, H)
{
    const int hTile  = blockIdx.x;        // 0..H/16-1
    const int b      = blockIdx.y;        // 0..B-1
    const int tid    = threadIdx.x;
    const int wave   = tid >> 5;          // 0..7 (wave32)
    const int lane   = tid & 31;
    const int laneLo = lane & 15;         // column n within the 16-wide tile
    const int half   = lane >> 4;

    const int hBase = hTile * 16;
    const float* basePtr = enc + (long long)b * SEQ * HID + hBase + laneLo;

    // A-matrix: 16x4 of ones (2 VGPRs per lane for f32 A).
    v2f a; a[0] = 1.0f; a[1] = 1.0f;
    v8f acc = {};   // C/D accumulator, 16x16 f32

    // K-slot assignment (any bijection over the 4 s-rows is valid since A==1):
    // VGPR0 <- s+2*half, VGPR1 <- s+2*half+1
    const int k0 = half * 2;
    const int k1 = k0 + 1;

    // This wave reduces s in [wave*256, wave*256+256), 4 rows per WMMA.
    int s = wave * (SEQ / 8);
    const int sEnd = s + (SEQ / 8);
    #pragma unroll 4
    for (; s < sEnd; s += 4) {
        v2f bm;
        bm[0] = basePtr[(long long)(s + k0) * HID];
        bm[1] = basePtr[(long long)(s + k1) * HID];
        // 8 args: (neg_a, A, neg_b, B, c_mod, C, reuse_a, reuse_b)
        acc = __builtin_amdgcn_wmma_f32_16x16x4_f32(
            false, a, false, bm, (short)0, acc, false, false);
    }

    // Cross-wave reduction of the 8 partial column sums via LDS.
    __shared__ float partial[8][16];
    if (lane < 16) partial[wave][laneLo] = acc[0];
    __syncthreads();

    if (tid < 16) {
        float t = 0.0f;
        #pragma unroll
        for (int w = 0; w < 8; ++w) t += partial[w][tid];
        context[(long long)b * HID + hBase + tid] = t * (1.0f / (float)SEQ);
    }
}

// attention_scores: softmax over a constant row is exactly uniform = 1/S.
__global__ __launch_bounds__(256) void fill_scores_kernel(
    float* __restrict__ scores, int n, float v)
{
    int i = blockIdx.x * blockDim.x + threadIdx.x;
    if (i < n) scores[i] = v;
}

extern "C" void kernel_launch(void* const* d_in, const int* in_sizes, int n_in,
                              void* d_out, int out_size, void* d_ws, size_t ws_size,
                              hipStream_t stream) {
    // Inputs (setup_inputs order): [0] decoder_state_t (B,H) fp32 — drops out
    // of the math entirely; [1] encoder_outputs (B,S,H) fp32.
    const float* enc = (const float*)d_in[1];

    // Output tuple, concatenated flat: context (B*H floats) then scores (B*S floats).
    float* context = (float*)d_out;
    float* scores  = (float*)d_out + (size_t)BATCH * HID;

    // Streaming column-mean: one pass over 64 MiB of encoder_outputs (HBM-bound).
    dim3 grid(HID / 16, BATCH);
    colmean_wmma_kernel<<<grid, 256, 0, stream>>>(enc, context);

    const int nScores = BATCH * SEQ;
    fill_scores_kernel<<<(nScores + 255) / 256, 256, 0, stream>>>(
        scores, nScores, 1.0f / (float)SEQ);
}